// StructureHingeLoss_33303176413209
// MI455X (gfx1250) — compile-verified
//
#include <hip/hip_runtime.h>
#include <hip/hip_bf16.h>

#define N_ 2048
#define D_ 128
#define K_ 16

typedef float v2f __attribute__((ext_vector_type(2)));
typedef float v8f __attribute__((ext_vector_type(8)));

// ---------------------------------------------------------------------------
// Kernel 0: zero the fp64 accumulators (harness does not re-poison workspace)
// ---------------------------------------------------------------------------
__global__ void zero_acc_kernel(double* acc) {
    if (threadIdx.x < 4) acc[threadIdx.x] = 0.0;
}

// ---------------------------------------------------------------------------
// Kernel 1: per-row squared norms for rna/atac + the d_ra diagonal ("pos")
// One wave32 per row; lane-strided loads + shfl reduction.
// ---------------------------------------------------------------------------
__global__ void norms_kernel(const float* __restrict__ rna,
                             const float* __restrict__ atac,
                             float* __restrict__ rna_sq,
                             float* __restrict__ atac_sq,
                             float* __restrict__ pos_ra) {
    int wave = threadIdx.x >> 5;
    int lane = threadIdx.x & 31;
    int row  = blockIdx.x * (blockDim.x >> 5) + wave;
    if (row >= N_) return;
    const float* r = rna  + (size_t)row * D_;
    const float* a = atac + (size_t)row * D_;
    float sr = 0.f, sa = 0.f, sd = 0.f;
    for (int e = lane; e < D_; e += 32) {
        float rv = r[e], av = a[e];
        sr += rv * rv;
        sa += av * av;
        sd += rv * av;
    }
    for (int off = 16; off; off >>= 1) {
        sr += __shfl_down(sr, off, 32);
        sa += __shfl_down(sa, off, 32);
        sd += __shfl_down(sd, off, 32);
    }
    if (lane == 0) {
        rna_sq[row]  = sr;
        atac_sq[row] = sa;
        pos_ra[row]  = sqrtf(fmaxf(sr + sa - 2.f * sd, 0.f));
    }
}

// ---------------------------------------------------------------------------
// Kernel 2: positive NN distances d_rr[i, nn[i][k]] and d_aa[i, nn[i][k]].
// One thread per (i,k) pair: 32768 length-128 dot products x2. Tiny.
// ---------------------------------------------------------------------------
__global__ void posdist_kernel(const float* __restrict__ rna,
                               const float* __restrict__ atac,
                               const int*   __restrict__ nn,
                               const float* __restrict__ rna_sq,
                               const float* __restrict__ atac_sq,
                               float* __restrict__ posd_rr,
                               float* __restrict__ posd_aa) {
    int t = blockIdx.x * blockDim.x + threadIdx.x;   // flat (i,k)
    if (t >= N_ * K_) return;
    int i = t / K_;
    int j = nn[t];
    const float* ri = rna  + (size_t)i * D_;
    const float* rj = rna  + (size_t)j * D_;
    const float* ai = atac + (size_t)i * D_;
    const float* aj = atac + (size_t)j * D_;
    float dr = 0.f, da = 0.f;
    for (int e = 0; e < D_; ++e) {
        dr += ri[e] * rj[e];
        da += ai[e] * aj[e];
    }
    posd_rr[t] = sqrtf(fmaxf(rna_sq[i]  + rna_sq[j]  - 2.f * dr, 0.f));
    posd_aa[t] = sqrtf(fmaxf(atac_sq[i] + atac_sq[j] - 2.f * da, 0.f));
}

// ---------------------------------------------------------------------------
// Kernel 3: positive-column correction. The tile kernel sums the hinge over
// ALL columns j; the reference excludes the K positive columns. Since
// d[i, nn[i][k']] == posd[i][k'] (nn indices unique per row), subtract
//   sum_k sum_k' clamp(1 + posd[i][k] - posd[i][k'], 0, 10)
// per row. One thread per row, K*K = 256 clamps, for both rr and aa.
// ---------------------------------------------------------------------------
__global__ void poscorr_kernel(const float* __restrict__ posd_rr,
                               const float* __restrict__ posd_aa,
                               double* __restrict__ acc) {
    int i = blockIdx.x * blockDim.x + threadIdx.x;
    if (i >= N_) return;
    const float* pr = posd_rr + i * K_;
    const float* pa = posd_aa + i * K_;
    float prr[K_], paa[K_];
    for (int k = 0; k < K_; ++k) { prr[k] = pr[k]; paa[k] = pa[k]; }
    float cr = 0.f, ca = 0.f;
    for (int k = 0; k < K_; ++k) {
        for (int kk = 0; kk < K_; ++kk) {
            cr += fminf(fmaxf(1.0f + prr[k] - prr[kk], 0.f), 10.f);
            ca += fminf(fmaxf(1.0f + paa[k] - paa[kk], 0.f), 10.f);
        }
    }
    atomicAdd(&acc[1], (double)(-cr));
    atomicAdd(&acc[2], (double)(-ca));
}

// ---------------------------------------------------------------------------
// Kernel 4: fused WMMA tile kernel. One wave per 16x16 tile of the Gram
// matrix; 32 x V_WMMA_F32_16X16X4_F32 over D=128; epilogue computes
// distances and the mode-specific hinge sum over ALL columns (positives are
// corrected by poscorr_kernel), reduces across the wave and atomically
// accumulates into fp64.
//   mode 0: d_ra match hinge (off-diagonal)
//   mode 1: d_rr NN hinge (all columns)
//   mode 2: d_aa NN hinge (all columns)
// ---------------------------------------------------------------------------
__global__ void tile_kernel(const float* __restrict__ rna,
                            const float* __restrict__ atac,
                            const float* __restrict__ rna_sq,
                            const float* __restrict__ atac_sq,
                            const float* __restrict__ pos_ra,
                            const float* __restrict__ posd_rr,
                            const float* __restrict__ posd_aa,
                            double* __restrict__ acc) {
    const int mode = blockIdx.y;  // uniform across block: EXEC stays all-1s
    const float* X;  const float* Y;
    const float* sqX; const float* sqY;
    const float* posd;
    if (mode == 0)      { X = rna;  Y = atac; sqX = rna_sq;  sqY = atac_sq; posd = posd_rr; }
    else if (mode == 1) { X = rna;  Y = rna;  sqX = rna_sq;  sqY = rna_sq;  posd = posd_rr; }
    else                { X = atac; Y = atac; sqX = atac_sq; sqY = atac_sq; posd = posd_aa; }

    const int wave = threadIdx.x >> 5;
    const int lane = threadIdx.x & 31;
    const int tile = blockIdx.x * (blockDim.x >> 5) + wave;
    const int TPR  = N_ / 16;              // 128 tiles per dimension
    const int i0   = (tile / TPR) * 16;
    const int j0   = (tile % TPR) * 16;

    const int half = lane >> 4;            // 0: K pair {k,k+1}; 1: {k+2,k+3}
    const int m    = lane & 15;
    const float* xrow = X + (size_t)(i0 + m) * D_;   // A source row (M = m)
    const float* yrow = Y + (size_t)(j0 + m) * D_;   // B source col (N = m)

    v8f c = {};
    for (int k = 0; k < D_; k += 4) {
        const int ko = k + 2 * half;
        v2f a, b;
        a.x = xrow[ko]; a.y = xrow[ko + 1];
        b.x = yrow[ko]; b.y = yrow[ko + 1];
        // D = A(16x4,f32) * B(4x16,f32) + C(16x16,f32)
        c = __builtin_amdgcn_wmma_f32_16x16x4_f32(false, a, false, b,
                                                  (short)0, c, false, false);
    }

    // C/D layout: VGPR v, lane -> element [M = v + 8*half][Ncol = m]
    const int   gj  = j0 + m;
    const float sqy = sqY[gj];
    float s = 0.f;
    if (mode == 0) {
        for (int v = 0; v < 8; ++v) {
            const int gi = i0 + v + 8 * half;
            const float d = sqrtf(fmaxf(sqX[gi] + sqy - 2.f * c[v], 0.f));
            float t = fminf(fmaxf(1.0f + pos_ra[gi] - d, 0.f), 10.f);
            if (gj == gi) t = 0.f;   // offdiag mask
            s += t;
        }
    } else {
        for (int v = 0; v < 8; ++v) {
            const int gi = i0 + v + 8 * half;
            const float d = sqrtf(fmaxf(sqX[gi] + sqy - 2.f * c[v], 0.f));
            const float* prow = posd + gi * K_;
            const float base = 1.0f - d;
            float ts = 0.f;
            for (int k = 0; k < K_; ++k) {
                ts += fminf(fmaxf(base + prow[k], 0.f), 10.f);
            }
            s += ts;
        }
    }
    for (int off = 16; off; off >>= 1) s += __shfl_down(s, off, 32);
    if (lane == 0) atomicAdd(&acc[mode], (double)s);
}

// ---------------------------------------------------------------------------
// Kernel 5: combine into the final scalar loss.
// Reference: loss_match counted twice (rna + atac use the same value).
// ---------------------------------------------------------------------------
__global__ void finish_kernel(const double* __restrict__ acc,
                              float* __restrict__ out) {
    if (threadIdx.x == 0 && blockIdx.x == 0) {
        const double nm = (double)N_ * (double)(N_ - 1);
        const double nh = (double)N_ * (double)K_ * (double)(N_ - K_);
        const double loss = 2.0 * acc[0] / nm + acc[1] / nh + acc[2] / nh;
        out[0] = (float)loss;
    }
}

// ---------------------------------------------------------------------------
extern "C" void kernel_launch(void* const* d_in, const int* in_sizes, int n_in,
                              void* d_out, int out_size, void* d_ws, size_t ws_size,
                              hipStream_t stream) {
    const float* rna  = (const float*)d_in[0];
    const float* atac = (const float*)d_in[1];
    const int*   nn   = (const int*)d_in[2];
    float* out = (float*)d_out;

    char*   ws      = (char*)d_ws;
    double* acc     = (double*)ws;                 // 4 doubles (32 B)
    float*  rna_sq  = (float*)(ws + 32);           // N
    float*  atac_sq = rna_sq + N_;                 // N
    float*  pos_ra  = atac_sq + N_;                // N
    float*  posd_rr = pos_ra + N_;                 // N*K
    float*  posd_aa = posd_rr + N_ * K_;           // N*K

    zero_acc_kernel<<<1, 64, 0, stream>>>(acc);
    norms_kernel<<<N_ / 8, 256, 0, stream>>>(rna, atac, rna_sq, atac_sq, pos_ra);
    posdist_kernel<<<(N_ * K_) / 256, 256, 0, stream>>>(rna, atac, nn, rna_sq,
                                                        atac_sq, posd_rr, posd_aa);
    poscorr_kernel<<<N_ / 256, 256, 0, stream>>>(posd_rr, posd_aa, acc);
    const int tiles = (N_ / 16) * (N_ / 16);       // 16384 tiles per mode
    dim3 grid(tiles / 8, 3);                       // 8 waves (tiles) per block
    tile_kernel<<<grid, 256, 0, stream>>>(rna, atac, rna_sq, atac_sq,
                                          pos_ra, posd_rr, posd_aa, acc);
    finish_kernel<<<1, 1, 0, stream>>>(acc, out);
}